// CNLoss_83193516524010
// MI455X (gfx1250) — compile-verified
//
#include <hip/hip_runtime.h>
#include <math.h>

#define BATCH   16384
#define NF      1000
#define NCLS    1000
#define NT      63        // number of 16-wide class/feature tiles (63*16 = 1008)
#define PAD     1008
#define LSTRIDE (BATCH + 64)   // per-tile list stride: room for 8+ sentinel entries
#define ALPHA_C 0.5f
#define BETA_C  0.003f
#define GAMMA_C 0.001f

typedef __attribute__((ext_vector_type(2))) float v2f;
typedef __attribute__((ext_vector_type(8))) float v8f;

__device__ __forceinline__ float wave_sum(float v) {
  for (int i = 16; i >= 1; i >>= 1) v += __shfl_xor(v, i, 32);
  return v;
}
__device__ __forceinline__ float wave_max(float v) {
  for (int i = 16; i >= 1; i >>= 1) v = fmaxf(v, __shfl_xor(v, i, 32));
  return v;
}

// ---------------- Kernel A: per-row stats (LSE, x[b,y], ||x||^2) -------------
// One wave32 per row; row (1000 f32 = 250 float4) held in registers.
__global__ void cn_row_stats(const float* __restrict__ x, const int* __restrict__ y,
                             float* __restrict__ ce_term, float* __restrict__ rowsq) {
  int wave = threadIdx.x >> 5;
  int lane = threadIdx.x & 31;
  int r = blockIdx.x * 8 + wave;
  if (r >= BATCH) return;
  const float4* row = (const float4*)(x + (size_t)r * NF);  // 1000/4 = 250, 16B aligned
  float4 v[8];
  int n = 0;
  float mx = -3.0e38f, sq = 0.f;
  for (int i = lane; i < 250; i += 32) {
    float4 t = row[i];
    v[n++] = t;
    mx = fmaxf(mx, fmaxf(fmaxf(t.x, t.y), fmaxf(t.z, t.w)));
    sq += t.x * t.x + t.y * t.y + t.z * t.z + t.w * t.w;
  }
  mx = wave_max(mx);
  float es = 0.f;
  for (int k = 0; k < n; ++k)
    es += __expf(v[k].x - mx) + __expf(v[k].y - mx) + __expf(v[k].z - mx) + __expf(v[k].w - mx);
  es = wave_sum(es);
  sq = wave_sum(sq);
  if (lane == 0) {
    float lse = mx + __logf(es);
    float xy = x[(size_t)r * NF + y[r]];
    ce_term[r] = lse - xy;   // -logp[b, y_b]
    rowsq[r]   = sq;
  }
}

// ------- Kernel B: deterministic ordered row lists per 16-class tile ---------
// One wave per class tile; ballot + prefix popcount keeps batch order stable.
// Entries are packed (row << 4) | (y - c0); 8 sentinel (-1) entries appended
// so the WMMA kernel can read unconditionally past the end.
__global__ void cn_build_lists(const int* __restrict__ y, int* __restrict__ list,
                               int* __restrict__ listcnt, int* __restrict__ counts) {
  int ct = blockIdx.x;
  int lane = threadIdx.x;
  __shared__ int cnt16[16];
  if (lane < 16) cnt16[lane] = 0;
  __syncthreads();
  int* lp = list + (size_t)ct * LSTRIDE;
  int base = 0;
  int c0 = ct << 4;
  for (int b = 0; b < BATCH; b += 32) {
    int yv = y[b + lane];
    bool pred = (yv >= c0) && (yv < c0 + 16);
    unsigned m = __builtin_amdgcn_ballot_w32(pred);
    if (pred) {
      int pos = base + __builtin_popcount(m & ((1u << lane) - 1u));
      lp[pos] = ((b + lane) << 4) | (yv - c0);
      atomicAdd(&cnt16[yv - c0], 1);   // integer atomic: order-independent value
    }
    base += __builtin_popcount(m);
  }
  if (lane < 8) lp[base + lane] = -1;  // sentinel pad (>= 2 chunks)
  __syncthreads();
  if (lane == 0) listcnt[ct] = base;
  if (lane < 16) counts[c0 + lane] = cnt16[lane];
}

// ------ Kernel C: sx = onehot(y)^T @ x via V_WMMA_F32_16X16X4_F32 ------------
// One wave owns one (class-tile, feature-tile) 16x16 output; K walks the
// packed row list in fixed order: branchless (cndmask-only), depth-1 pipelined
// index loads, unconditional x loads. Deterministic, single pass over x.
__global__ void cn_segsum_wmma(const float* __restrict__ x,
                               const int* __restrict__ list, const int* __restrict__ listcnt,
                               float* __restrict__ sx) {
  int ct = blockIdx.x;
  int wave = threadIdx.x >> 5;
  int lane = threadIdx.x & 31;
  int ft = blockIdx.y * 4 + wave;
  if (ft >= NT) return;                 // whole-wave uniform exit; EXEC stays full
  int half = lane >> 4;
  int nn = lane & 15;
  int c0 = ct << 4;
  int f0 = ft << 4;
  int fidx = f0 + nn; if (fidx > NF - 1) fidx = NF - 1;   // padded cols never read back
  int cnt = listcnt[ct];
  int cnt4 = (cnt + 3) & ~3;
  const int* lp = list + (size_t)ct * LSTRIDE;
  v8f acc = {0.f, 0.f, 0.f, 0.f, 0.f, 0.f, 0.f, 0.f};
  // lane layout: A(16x4 MxK) m=lane&15, k=2*(lane>>4)+vgpr ; B(4x16 KxN) same k, n=lane&15
  int i0 = half << 1;         // my two K slots within a chunk: i0, i0+1
  int e0 = lp[i0];
  int e1 = lp[i0 + 1];
  for (int p = 0; p < cnt4; p += 4) {
    int ne0 = lp[p + 4 + i0];       // sentinel-padded: always in-bounds
    int ne1 = lp[p + 5 + i0];
    __builtin_prefetch(lp + p + 16, 0, 3);   // global_prefetch_b8
    int rr0 = (e0 >= 0) ? (e0 >> 4) : 0;     // sentinel -> row 0 (valid addr)
    int rr1 = (e1 >= 0) ? (e1 >> 4) : 0;
    v2f b, a;
    b.x = x[(size_t)rr0 * NF + fidx];
    b.y = x[(size_t)rr1 * NF + fidx];
    a.x = (e0 >= 0 && (e0 & 15) == nn) ? 1.f : 0.f;
    a.y = (e1 >= 0 && (e1 & 15) == nn) ? 1.f : 0.f;
    acc = __builtin_amdgcn_wmma_f32_16x16x4_f32(
        false, a, false, b, (short)0, acc, false, false);
    e0 = ne0;
    e1 = ne1;
  }
  // C/D layout: vgpr r -> m = r + 8*half, n = lane&15
  for (int r = 0; r < 8; ++r) {
    int m = (half << 3) + r;
    sx[(size_t)(c0 + m) * PAD + (f0 + nn)] = acc[r];
  }
}

// ---- Kernel E: per-class new centers + 5 per-class scalar reductions --------
__global__ void cn_per_class(const float* __restrict__ centers, const float* __restrict__ sx,
                             const int* __restrict__ counts,
                             const int* __restrict__ list, const int* __restrict__ listcnt,
                             const float* __restrict__ rowsq,
                             float* __restrict__ nc, float* __restrict__ cls) {
  int c = blockIdx.x;
  int t = threadIdx.x;   // 128
  int cnt = counts[c];
  float scale = (cnt > 0) ? (ALPHA_C / (float)cnt) : 0.f;
  float keep  = (cnt > 0) ? (1.f - ALPHA_C) : 1.f;
  float dA = 0.f, dB = 0.f, nO = 0.f, nN = 0.f;
  const float* crow = centers + (size_t)c * NF;
  const float* srow = sx + (size_t)c * PAD;
  float* nrow = nc + (size_t)c * NF;
  for (int f = t; f < NF; f += 128) {
    float cv = crow[f];
    float sv = srow[f];
    float nv = keep * cv + scale * sv;   // centers + alpha*(sx/cnt - centers)
    nrow[f] = nv;
    dA += sv * cv;
    dB += sv * nv;
    nO += cv * cv;
    nN += nv * nv;
  }
  float sq = 0.f;
  int ct = c >> 4;
  int sub = c & 15;
  int lc = listcnt[ct];
  const int* lp = list + (size_t)ct * LSTRIDE;
  for (int i = t; i < lc; i += 128) {
    int e = lp[i];                       // packed (row<<4)|(y-c0), e >= 0 here
    if ((e & 15) == sub) sq += rowsq[e >> 4];
  }
  __shared__ float red[5][128];
  red[0][t] = dA; red[1][t] = dB; red[2][t] = nO; red[3][t] = nN; red[4][t] = sq;
  __syncthreads();
  for (int s = 64; s >= 1; s >>= 1) {
    if (t < s)
      for (int k = 0; k < 5; ++k) red[k][t] += red[k][t + s];
    __syncthreads();
  }
  if (t == 0) {
    cls[0 * NCLS + c] = red[0][0];  // <sx, centers>
    cls[1 * NCLS + c] = red[1][0];  // <sx, nc>
    cls[2 * NCLS + c] = red[2][0];  // ||centers||^2
    cls[3 * NCLS + c] = red[3][0];  // ||nc||^2
    cls[4 * NCLS + c] = red[4][0];  // sum_{y=c} ||x_b||^2
  }
}

// ---------------- Kernel F: column sums of new centers -----------------------
__global__ void cn_col_sums(const float* __restrict__ nc, float* __restrict__ colsum) {
  int f = blockIdx.x * 256 + threadIdx.x;
  if (f >= NF) return;
  float s = 0.f;
  for (int c = 0; c < NCLS; ++c) s += nc[(size_t)c * NF + f];
  colsum[f] = s;
}

// ---------------- Kernel G: final deterministic combine ----------------------
__device__ __forceinline__ float block_reduce_1024(float v, float* red, int t) {
  red[t] = v; __syncthreads();
  for (int s = 512; s >= 1; s >>= 1) {
    if (t < s) red[t] += red[t + s];
    __syncthreads();
  }
  float r = red[0]; __syncthreads();
  return r;
}

__global__ void cn_finalize(const float* __restrict__ ce_term, const float* __restrict__ rowsq,
                            const float* __restrict__ cls, const int* __restrict__ counts,
                            const float* __restrict__ colsum, float* __restrict__ out) {
  int t = threadIdx.x;   // 1024
  __shared__ float red[1024];
  float pCe = 0.f, pRsq = 0.f;
  for (int i = t; i < BATCH; i += 1024) { pCe += ce_term[i]; pRsq += rowsq[i]; }
  float pdA = 0.f, pCN2o = 0.f, pN2n = 0.f, pPcm = 0.f, pS2 = 0.f;
  for (int c = t; c < NCLS; c += 1024) {
    float cntf = (float)counts[c];
    pdA   += cls[0 * NCLS + c];
    pCN2o += cntf * cls[2 * NCLS + c];
    float n2n = cls[3 * NCLS + c];
    pN2n += n2n;
    if (counts[c] > 0) {
      float pcs = cls[4 * NCLS + c] - 2.f * cls[1 * NCLS + c] + cntf * n2n;
      pPcm += pcs / cntf;
    }
    float sv = colsum[c];   // NF == NCLS
    pS2 += sv * sv;
  }
  float sCe   = block_reduce_1024(pCe,  red, t);
  float sRsq  = block_reduce_1024(pRsq, red, t);
  float sdA   = block_reduce_1024(pdA,  red, t);
  float sCN2o = block_reduce_1024(pCN2o, red, t);
  float sN2n  = block_reduce_1024(pN2n, red, t);
  float sPcm  = block_reduce_1024(pPcm, red, t);
  float sS2   = block_reduce_1024(pS2,  red, t);
  if (t == 0) {
    float ce = sCe / (float)BATCH;
    float center_loss = (sRsq - 2.f * sdA + sCN2o) / (float)BATCH;
    float num_pairs = 0.5f * (float)NCLS * (float)(NCLS - 1);
    float inter = ((float)NCLS * sN2n - sS2) / num_pairs;
    float intra = sPcm / (float)NCLS;
    out[0] = ce + BETA_C * center_loss + GAMMA_C * inter + GAMMA_C * intra;
  }
}

extern "C" void kernel_launch(void* const* d_in, const int* in_sizes, int n_in,
                              void* d_out, int out_size, void* d_ws, size_t ws_size,
                              hipStream_t stream) {
  (void)in_sizes; (void)n_in; (void)out_size; (void)ws_size;
  const float* x       = (const float*)d_in[0];
  const int*   y       = (const int*)d_in[1];
  const float* centers = (const float*)d_in[2];
  float* out = (float*)d_out;

  char* w = (char*)d_ws;
  size_t off = 0;
  auto take = [&](size_t bytes) -> void* {
    void* p = (void*)(w + off);
    off += (bytes + 255) & ~(size_t)255;
    return p;
  };
  float* sx      = (float*)take((size_t)PAD * PAD * 4);      // 1008x1008 padded
  float* nc      = (float*)take((size_t)NCLS * NF * 4);      // updated centers
  int*   list    = (int*)  take((size_t)NT * LSTRIDE * 4);   // ordered packed row lists
  float* ce_term = (float*)take((size_t)BATCH * 4);
  float* rowsq   = (float*)take((size_t)BATCH * 4);
  int*   listcnt = (int*)  take((size_t)NT * 4);
  int*   counts  = (int*)  take((size_t)PAD * 4);
  float* cls     = (float*)take((size_t)5 * NCLS * 4);
  float* colsum  = (float*)take((size_t)NF * 4);

  cn_row_stats  <<<BATCH / 8, 256, 0, stream>>>(x, y, ce_term, rowsq);
  cn_build_lists<<<NT, 32, 0, stream>>>(y, list, listcnt, counts);
  cn_segsum_wmma<<<dim3(NT, 16), 128, 0, stream>>>(x, list, listcnt, sx);
  cn_per_class  <<<NCLS, 128, 0, stream>>>(centers, sx, counts, list, listcnt,
                                           rowsq, nc, cls);
  cn_col_sums   <<<4, 256, 0, stream>>>(nc, colsum);
  cn_finalize   <<<1, 1024, 0, stream>>>(ce_term, rowsq, cls, counts, colsum, out);
}